// HistogramLoss_56221121904724
// MI455X (gfx1250) — compile-verified
//
#include <hip/hip_runtime.h>

typedef __attribute__((ext_vector_type(2))) float v2f;
typedef __attribute__((ext_vector_type(8))) float v8f;

#define NUM_STEPS 151
#define INV_STEP  75.0f          // 1/STEP, STEP = 2/(151-1); exact in f32
#define FEAT_D    256
#define NWAVES    8              // 256 threads = 8 wave32
#define DUMMY_BIN 151            // clamp target for dropped contributions (never flushed)

// ---------------- kernel 0: zero workspace ----------------
__global__ void hl_zero_ws(float* ws, int n) {
    int i = blockIdx.x * blockDim.x + threadIdx.x;
    if (i < n) ws[i] = 0.0f;
}

// ---------------- kernel 1: fused gram + soft histogram ----------------
// Each wave owns one 16x16 tile of the Gram matrix at a time, computed with
// chained V_WMMA_F32_16X16X4_F32 (full fp32 precision), then scatters the 256
// similarities into wave-private LDS histograms via ds_add_f32.
__global__ __launch_bounds__(256) void hl_gram_hist(
    const float* __restrict__ feats, const int* __restrict__ labels, int N,
    float* __restrict__ gP, float* __restrict__ gN, unsigned* __restrict__ gCnt)
{
    __shared__ float    lhist[NWAVES][2][NUM_STEPS + 1];
    __shared__ unsigned lcnt[NWAVES];

    const int tid  = threadIdx.x;
    const int wave = tid >> 5;
    const int lane = tid & 31;

    for (int i = tid; i < NWAVES * 2 * (NUM_STEPS + 1); i += blockDim.x)
        ((float*)lhist)[i] = 0.0f;
    if (tid < NWAVES) lcnt[tid] = 0u;
    __syncthreads();

    const int ntb        = N >> 4;              // tiles per side
    const int numTiles   = ntb * ntb;
    const int wavesTotal = gridDim.x * (blockDim.x >> 5);
    const int gw         = blockIdx.x * (blockDim.x >> 5) + wave;

    // fp32 16x16x4 fragment layout: lane holds row (lane&15), K sub-pair (lane>>4)
    const int r  = lane & 15;
    const int hi = lane >> 4;

    float* hP = lhist[wave][0];
    float* hN = lhist[wave][1];
    unsigned myPos = 0;

    for (int t = gw; t < numTiles; t += wavesTotal) {
        int bi = t / ntb;
        int bj = t - bi * ntb;
        if (bi > bj) continue;                  // wave-uniform: EXEC stays all-1s

        const float* arow = feats + (size_t)(bi * 16 + r) * FEAT_D + (hi << 1);
        const float* brow = feats + (size_t)(bj * 16 + r) * FEAT_D + (hi << 1);

        v8f c = {};
        #pragma unroll 8
        for (int k = 0; k < FEAT_D / 4; ++k) {
            v2f a = *(const v2f*)(arow + 4 * k);
            v2f b = *(const v2f*)(brow + 4 * k);
            // D = A x B + C, fp32 in/out (V_WMMA_F32_16X16X4_F32)
            c = __builtin_amdgcn_wmma_f32_16x16x4_f32(
                    false, a, false, b, (short)0, c, false, false);
        }

        // C/D layout: VGPR v -> (M = v + 8*hi, N = lane&15)
        const int gj      = bj * 16 + (lane & 15);
        const int lj      = labels[gj];
        const int rowbase = bi * 16 + 8 * hi;   // multiple of 8 -> 32B aligned

        // 8 consecutive row labels via two b128 loads instead of 8 b32 loads
        int4 rl0 = *(const int4*)(labels + rowbase);
        int4 rl1 = *(const int4*)(labels + rowbase + 4);
        int rlab[8] = { rl0.x, rl0.y, rl0.z, rl0.w, rl1.x, rl1.y, rl1.z, rl1.w };

        #pragma unroll
        for (int v = 0; v < 8; ++v) {
            int gi = rowbase + v;
            if (gi < gj) {                       // strict upper triangle
                float s   = c[v];
                bool  pos = (rlab[v] == lj);
                // wa = (s - t_j)/STEP == frac(s*75);  b0 = floor(s*75) + 75
                float sf  = s * INV_STEP;
                float fl  = floorf(sf);
                float wa  = sf - fl;
                float wb  = 1.0f - wa;
                int   b0  = (int)fl + (NUM_STEPS - 1) / 2;
                // clamp dropped (out-of-range) bins to dummy slot; no exec juggling
                int ba = ((unsigned)b0       < (unsigned)NUM_STEPS) ? b0     : DUMMY_BIN;
                int bb = ((unsigned)(b0 + 1) < (unsigned)NUM_STEPS) ? b0 + 1 : DUMMY_BIN;
                float* h = pos ? hP : hN;
                atomicAdd(&h[ba], wa);
                atomicAdd(&h[bb], wb);
                myPos += pos ? 1u : 0u;
            }
        }
    }

    atomicAdd(&lcnt[wave], myPos);
    __syncthreads();

    // one flush per block: sum the 8 wave-private histograms, then global atomics
    for (int idx = tid; idx < NUM_STEPS; idx += blockDim.x) {
        float sp = 0.f, sn = 0.f;
        #pragma unroll
        for (int w = 0; w < NWAVES; ++w) { sp += lhist[w][0][idx]; sn += lhist[w][1][idx]; }
        if (sp != 0.f) atomicAdd(&gP[idx], sp);
        if (sn != 0.f) atomicAdd(&gN[idx], sn);
    }
    if (tid == 0) {
        unsigned tot = 0;
        #pragma unroll
        for (int w = 0; w < NWAVES; ++w) tot += lcnt[w];
        atomicAdd(gCnt, tot);
    }
}

// ---------------- kernel 2: finalize (normalize, cumsum, dot) ----------------
__global__ void hl_finalize(const float* __restrict__ gP, const float* __restrict__ gN,
                            const unsigned* __restrict__ gCnt, float* __restrict__ out, int N)
{
    if (threadIdx.x == 0 && blockIdx.x == 0) {
        double totalPairs = 0.5 * (double)N * (double)(N - 1);
        float ps = (float)(*gCnt);
        float ns = (float)totalPairs - ps;
        float cum = 0.0f, loss = 0.0f;
        for (int k = 0; k < NUM_STEPS; ++k) {
            cum  += gP[k] / ps;                 // cdf of h_pos
            loss += (gN[k] / ns) * cum;         // sum(h_neg * cdf_pos)
        }
        out[0] = loss;
    }
}

extern "C" void kernel_launch(void* const* d_in, const int* in_sizes, int n_in,
                              void* d_out, int out_size, void* d_ws, size_t ws_size,
                              hipStream_t stream) {
    const float* feats  = (const float*)d_in[0];
    const int*   labels = (const int*)d_in[1];
    const int    N      = in_sizes[1];          // 4096

    float*    ws   = (float*)d_ws;
    float*    gP   = ws;                         // [151] pos histogram (unnormalized)
    float*    gN   = ws + NUM_STEPS;             // [151] neg histogram (unnormalized)
    unsigned* gCnt = (unsigned*)(ws + 2 * NUM_STEPS);  // [1] positive-pair count

    hl_zero_ws<<<1, 512, 0, stream>>>(ws, 2 * NUM_STEPS + 1);

    hl_gram_hist<<<1024, 256, 0, stream>>>(feats, labels, N, gP, gN, gCnt);

    hl_finalize<<<1, 32, 0, stream>>>(gP, gN, gCnt, (float*)d_out, N);
}